// Evaluator_26302379721075
// MI455X (gfx1250) — compile-verified
//
#include <hip/hip_runtime.h>

typedef __attribute__((ext_vector_type(2))) float v2f;
typedef __attribute__((ext_vector_type(8))) float v8f;

// Wave32 sum-reduction via V_WMMA_F32_16X16X4_F32.
// Place each lane's partial s into A[m, k] with m = lane%16, k = (lane<16 ? 0 : 2)
// (A 16x4 f32 layout: lanes 0-15 hold K=0/1 in VGPR0/1, lanes 16-31 hold K=2/3).
// With B = ones(4x16), C = 0:  D[m,n] = s_m + s_{m+16} for every n.
// D layout: lane n (0-15) VGPR v = D[v, n]; lane n (16-31) VGPR v = D[v+8, n-16].
// So sum of my 8 D VGPRs = half-sum; adding the value from lane^16 gives the
// full 32-lane total in every lane.
__device__ __forceinline__ float wave_sum_wmma(float s) {
  v2f a; a[0] = s;    a[1] = 0.0f;
  v2f b; b[0] = 1.0f; b[1] = 1.0f;
  v8f c = {};
  v8f d = __builtin_amdgcn_wmma_f32_16x16x4_f32(
      /*neg_a=*/false, a, /*neg_b=*/false, b,
      /*c_mod=*/(short)0, c, /*reuse_a=*/false, /*reuse_b=*/false);
  float p = ((d[0] + d[1]) + (d[2] + d[3])) + ((d[4] + d[5]) + (d[6] + d[7]));
  p += __shfl_xor(p, 16, 32);
  return p;
}

__global__ void acc_kernel(const float* __restrict__ pred,
                           const int* __restrict__ label,
                           float* __restrict__ ws,
                           int num_slices, unsigned bsz) {
  const int tid    = blockIdx.x * blockDim.x + threadIdx.x;
  const int stride = gridDim.x * blockDim.x;

  float cnt = 0.0f;
  for (int s = tid; s < num_slices; s += stride) {
    const float* base = pred + (size_t)s * 10;   // 40B slice, 8B aligned
    // 5 x b64 non-temporal loads (stream > L2 capacity, bypass residency)
    v2f x0 = __builtin_nontemporal_load((const v2f*)(base + 0));
    v2f x1 = __builtin_nontemporal_load((const v2f*)(base + 2));
    v2f x2 = __builtin_nontemporal_load((const v2f*)(base + 4));
    v2f x3 = __builtin_nontemporal_load((const v2f*)(base + 6));
    v2f x4 = __builtin_nontemporal_load((const v2f*)(base + 8));

    float v[10] = {x0[0], x0[1], x1[0], x1[1], x2[0],
                   x2[1], x3[0], x3[1], x4[0], x4[1]};

    // first-occurrence argmax over the 10 class scores
    int   best = 0;
    float bv   = v[0];
#pragma unroll
    for (int ci = 1; ci < 10; ++ci) {
      if (v[ci] > bv) { bv = v[ci]; best = ci; }
    }

    const int lb = label[(unsigned)s % bsz];     // b = s mod B (labels L2-resident)
    cnt += (best == lb) ? 1.0f : 0.0f;
  }

  // Full EXEC here (structured reconvergence after the loop) -> WMMA legal.
  const float wave_total = wave_sum_wmma(cnt);
  if ((threadIdx.x & 31) == 0) {
    atomicAdd(ws, wave_total);                   // global_atomic_add_f32
  }
}

__global__ void finalize_kernel(const float* __restrict__ ws,
                                float* __restrict__ out, float inv_total) {
  out[0] = ws[0] * inv_total;
}

extern "C" void kernel_launch(void* const* d_in, const int* in_sizes, int n_in,
                              void* d_out, int out_size, void* d_ws, size_t ws_size,
                              hipStream_t stream) {
  const float* pred  = (const float*)d_in[0];
  const int*   label = (const int*)d_in[1];
  float*       ws    = (float*)d_ws;
  float*       out   = (float*)d_out;

  const int C          = 10;
  const int num_slices = in_sizes[0] / C;        // M*K*N*B = 8,388,608
  const unsigned bsz   = (unsigned)in_sizes[1];  // B = 2048

  hipMemsetAsync(ws, 0, sizeof(float), stream);  // capture-safe memset node

  const int threads = 256;                       // 8 waves / block
  const int blocks  = 2048;                      // 16 slices per thread
  acc_kernel<<<blocks, threads, 0, stream>>>(pred, label, ws, num_slices, bsz);
  finalize_kernel<<<1, 1, 0, stream>>>(ws, out, 1.0f / (float)num_slices);
}